// DiversityLoss_nojoint_2963527434409
// MI455X (gfx1250) — compile-verified
//
#include <hip/hip_runtime.h>
#include <hip/hip_bf16.h>
#include <math.h>

typedef __attribute__((ext_vector_type(2))) float v2f;
typedef __attribute__((ext_vector_type(8))) float v8f;

#define K_HYP 32
#define B_DIM 32
#define T_DIM 64
#define E_DIM 63
#define NGROUP (B_DIM * T_DIM)               // 2048 (b,t) groups
#define ROW_STRIDE (B_DIM * T_DIM * E_DIM)   // 129024 floats between k-rows
#define S_LDS 68                             // padded row stride in floats (272B, 16B aligned)
#define WAVES_PER_BLOCK 4
#define NPAIR (K_HYP * (K_HYP - 1) / 2)      // 496

#if defined(__has_builtin)
#  if __has_builtin(__builtin_amdgcn_global_load_async_to_lds_b32)
#    define HAVE_ASYNC_LDS 1
#  endif
#endif
#ifndef HAVE_ASYNC_LDS
#  define HAVE_ASYNC_LDS 0
#endif

typedef __attribute__((address_space(1))) int* gptr_t;   // printed as "__device__ int *"
typedef __attribute__((address_space(3))) int* lptr_t;   // printed as "__shared__ int *"

__global__ __launch_bounds__(WAVES_PER_BLOCK * 32)
void div_loss_main(const float* __restrict__ pose, float* __restrict__ partial) {
  __shared__ float tile[WAVES_PER_BLOCK * K_HYP * S_LDS];  // 34816 B

  const int lane = threadIdx.x & 31;
  const int wave = threadIdx.x >> 5;
  const int g = blockIdx.x * WAVES_PER_BLOCK + wave;       // (b,t) group, always < 2048

  float* wt = &tile[wave * K_HYP * S_LDS];
  const float* src = pose + g * E_DIM;

  // Stage the 32 x 63 tile for this group into LDS (rows padded to 68 floats).
  for (int m = lane; m < K_HYP * E_DIM; m += 32) {
    const int k = m / E_DIM;
    const int e = m - k * E_DIM;
#if HAVE_ASYNC_LDS
    __builtin_amdgcn_global_load_async_to_lds_b32(
        (gptr_t)(src + k * ROW_STRIDE + e),
        (lptr_t)(&wt[k * S_LDS + e]), 0, 0);
#else
    wt[k * S_LDS + e] = src[k * ROW_STRIDE + e];
#endif
  }
  // Zero the pad element e==63 so the vectorized 64-wide inner loop adds |0-0|.
  wt[lane * S_LDS + E_DIM] = 0.0f;

#if HAVE_ASYNC_LDS
#  if __has_builtin(__builtin_amdgcn_s_wait_asynccnt)
  __builtin_amdgcn_s_wait_asynccnt(0);
#  else
  asm volatile("s_wait_asynccnt 0" ::: "memory");
#  endif
#endif
  __syncthreads();

  // Each lane processes pair indices p = it*32 + lane over the 496-pair triangle.
  float acc = 0.0f;
  int i = 0, base = 0;  // base = number of pairs with first index < i
#pragma unroll 1
  for (int it = 0; it < 16; ++it) {
    const int p = it * 32 + lane;
    const bool valid = (p < NPAIR);
    const int pc = valid ? p : (NPAIR - 1);   // clamp keeps control flow uniform-ish
    while (i < K_HYP - 1 && pc >= base + (K_HYP - 1 - i)) { base += K_HYP - 1 - i; ++i; }
    const int j = i + 1 + (pc - base);

    const float4* ra = (const float4*)&wt[i * S_LDS];
    const float4* rb = (const float4*)&wt[j * S_LDS];
    float d = 0.0f;
#pragma unroll
    for (int c = 0; c < 16; ++c) {            // 64 floats (e=63 padded with 0)
      const float4 a = ra[c];
      const float4 b = rb[c];
      d += fabsf(a.x - b.x) + fabsf(a.y - b.y) + fabsf(a.z - b.z) + fabsf(a.w - b.w);
    }
    const float ev = __expf(-d);
    acc += valid ? ev : 0.0f;
  }

  // WMMA horizontal reduction (all lanes active; EXEC == all 1s here):
  // D = Ones(16x4) * B(4x16) + 0  => every row of D holds the column sums of B.
  // Each lane contributes acc into one B slot (other slot zero), so
  // sum over D row 0 == sum of all 32 lane partials, exactly in fp32.
  v2f a1 = {1.0f, 1.0f};
  v2f bp = {acc, 0.0f};
  v8f cz = {0.f, 0.f, 0.f, 0.f, 0.f, 0.f, 0.f, 0.f};
  v8f dm = __builtin_amdgcn_wmma_f32_16x16x4_f32(false, a1, false, bp,
                                                 (short)0, cz, false, false);
  float t = dm[0];                 // row M=0 (lanes 0..15) / M=8 (lanes 16..31): same colsums
  t += __shfl_xor(t, 1, 32);       // sum the 16 colsums within each 16-lane half
  t += __shfl_xor(t, 2, 32);
  t += __shfl_xor(t, 4, 32);
  t += __shfl_xor(t, 8, 32);

  if (lane == 0) partial[g] = t;
}

__global__ __launch_bounds__(256)
void div_loss_reduce(const float* __restrict__ partial, float* __restrict__ out) {
  __shared__ float wsum[8];
  float s = 0.0f;
  for (int idx = threadIdx.x; idx < NGROUP; idx += 256) s += partial[idx];
  s += __shfl_xor(s, 16, 32);
  s += __shfl_xor(s, 8, 32);
  s += __shfl_xor(s, 4, 32);
  s += __shfl_xor(s, 2, 32);
  s += __shfl_xor(s, 1, 32);
  if ((threadIdx.x & 31) == 0) wsum[threadIdx.x >> 5] = s;
  __syncthreads();
  if (threadIdx.x == 0) {
    float t = 0.0f;
#pragma unroll
    for (int w = 0; w < 8; ++w) t += wsum[w];
    out[0] = t * (1.0f / (float)((long long)NGROUP * NPAIR));
  }
}

extern "C" void kernel_launch(void* const* d_in, const int* in_sizes, int n_in,
                              void* d_out, int out_size, void* d_ws, size_t ws_size,
                              hipStream_t stream) {
  (void)in_sizes; (void)n_in; (void)out_size; (void)ws_size;
  const float* pose = (const float*)d_in[0];
  float* out = (float*)d_out;
  float* partial = (float*)d_ws;   // 2048 floats of scratch

  div_loss_main<<<NGROUP / WAVES_PER_BLOCK, WAVES_PER_BLOCK * 32, 0, stream>>>(pose, partial);
  div_loss_reduce<<<1, 256, 0, stream>>>(partial, out);
}